// SearcherDecoder_38379827757684
// MI455X (gfx1250) — compile-verified
//
#include <hip/hip_runtime.h>
#include <math.h>

typedef float v2f __attribute__((ext_vector_type(2)));
typedef float v8f __attribute__((ext_vector_type(8)));

#define B_  32
#define E_  1024
#define H_  1024
#define V_  32000
#define L_  64

// D = A(16x4 f32) x B(4x16 f32) + C(16x16 f32), wave32 WMMA
__device__ __forceinline__ v8f wmma_f32_k4(v2f a, v2f b, v8f c) {
    return __builtin_amdgcn_wmma_f32_16x16x4_f32(
        /*neg_a=*/false, a, /*neg_b=*/false, b,
        /*c_mod=*/(short)0, c, /*reuse_a=*/false, /*reuse_b=*/false);
}

// CDNA5 async copy: 16B per lane, global -> LDS, tracked by ASYNCcnt.
__device__ __forceinline__ void async_copy_b128(unsigned lds_byte_addr,
                                                const void* gaddr) {
    asm volatile("global_load_async_to_lds_b128 %0, %1, off"
                 :: "v"(lds_byte_addr), "v"((unsigned long long)gaddr)
                 : "memory");
}
__device__ __forceinline__ void wait_async_zero() {
    asm volatile("s_wait_asynccnt 0x0" ::: "memory");
}

__device__ __forceinline__ float fast_sigmoid(float x) {
    return 1.0f / (1.0f + __expf(-x));
}
__device__ __forceinline__ float fast_tanh(float x) {
    return 1.0f - 2.0f / (__expf(2.0f * x) + 1.0f);
}

// -------- init: h0 = 0, tok0 = dec_input --------
__global__ void init_kernel(const int* __restrict__ dec_input,
                            float* __restrict__ h0, int* __restrict__ tok0) {
    int i = blockIdx.x * blockDim.x + threadIdx.x;
    if (i < B_ * H_) h0[i] = 0.0f;
    if (i < B_) tok0[i] = dec_input[i];
}

// -------- fused embedding-gather + GRU cell --------
// 4 waves/block; each wave owns a 16-column n-tile for all 32 batch rows.
// h is staged to LDS once per block via async-to-LDS; x (embeddings) stream
// direct from L2 (per-lane gather addresses).
__global__ __launch_bounds__(128) void gru_step_kernel(
    const int*   __restrict__ tok,    // (B)
    const float* __restrict__ embed,  // (V,E)
    const float* __restrict__ W_ih,   // (3H,E)
    const float* __restrict__ W_hh,   // (3H,H)
    const float* __restrict__ b_ih,   // (3H)
    const float* __restrict__ b_hh,   // (3H)
    const float* __restrict__ h_in,   // (B,H)
    float*       __restrict__ h_out)  // (B,H)
{
    __shared__ float shh[B_ * H_];    // 128 KB staged h
    const int tid = threadIdx.x;

    {   // ---- async stage h (B x H) into LDS ----
        const unsigned lbase = (unsigned)(size_t)(void*)shh;
#pragma unroll 1
        for (int i = 0; i < (B_ * H_) / (128 * 4); ++i) {   // 64 iters
            const int f = (i * 128 + tid) * 4;              // float index
            async_copy_b128(lbase + (unsigned)f * 4u, (const void*)(h_in + f));
        }
        wait_async_zero();
        __syncthreads();
    }

    const int wave = tid >> 5;
    const int lane = tid & 31;
    const int half = lane >> 4;      // 0: lanes 0-15, 1: lanes 16-31
    const int mlo  = lane & 15;
    const int koff = half * 2;       // A/B fragment K offset per ISA layout
    const int n0   = (blockIdx.x * 4 + wave) * 16;   // column tile in [0,H)

    // A-matrix row pointers: x rows (embedding gather, global) and h rows (LDS)
    const long t0 = tok[mlo];
    const long t1 = tok[16 + mlo];
    const float* px0 = embed + (size_t)t0 * E_ + koff;
    const float* px1 = embed + (size_t)t1 * E_ + koff;
    const float* ph0 = shh + (size_t)mlo * H_ + koff;
    const float* ph1 = shh + (size_t)(16 + mlo) * H_ + koff;

    // B-matrix row pointers: W[g*H + n0 + n][k], n = lane%16
    const float* pwx[3];
    const float* pwh[3];
#pragma unroll
    for (int g = 0; g < 3; ++g) {
        pwx[g] = W_ih + (size_t)(g * H_ + n0 + mlo) * E_ + koff;
        pwh[g] = W_hh + (size_t)(g * H_ + n0 + mlo) * H_ + koff;
    }

    v8f ax[3][2] = {};   // gx accumulators (r,z,n) x (m-tile 0/1)
    v8f ah[3][2] = {};   // gh accumulators

#pragma unroll 1
    for (int k = 0; k < E_; k += 4) {
        v2f a_x0 = *(const v2f*)(px0 + k);
        v2f a_x1 = *(const v2f*)(px1 + k);
        v2f a_h0 = *(const v2f*)(ph0 + k);   // ds_load_b64
        v2f a_h1 = *(const v2f*)(ph1 + k);
#pragma unroll
        for (int g = 0; g < 3; ++g) {
            v2f bx = *(const v2f*)(pwx[g] + k);
            v2f bh = *(const v2f*)(pwh[g] + k);
            ax[g][0] = wmma_f32_k4(a_x0, bx, ax[g][0]);
            ax[g][1] = wmma_f32_k4(a_x1, bx, ax[g][1]);
            ah[g][0] = wmma_f32_k4(a_h0, bh, ah[g][0]);
            ah[g][1] = wmma_f32_k4(a_h1, bh, ah[g][1]);
        }
    }

    const int col = n0 + mlo;
    const float bir  = b_ih[col],           bhr = b_hh[col];
    const float biz  = b_ih[H_ + col],      bhz = b_hh[H_ + col];
    const float bin_ = b_ih[2 * H_ + col],  bhn = b_hh[2 * H_ + col];

#pragma unroll
    for (int tm = 0; tm < 2; ++tm) {
#pragma unroll
        for (int i = 0; i < 8; ++i) {
            const int m = tm * 16 + half * 8 + i;   // C/D layout: m = 8*half + i (+16)
            float xr = ax[0][tm][i] + bir, hr = ah[0][tm][i] + bhr;
            float xz = ax[1][tm][i] + biz, hz = ah[1][tm][i] + bhz;
            float xn = ax[2][tm][i] + bin_, hn = ah[2][tm][i] + bhn;
            float r = fast_sigmoid(xr + hr);
            float z = fast_sigmoid(xz + hz);
            float n = fast_tanh(xn + r * hn);
            float hp = shh[(size_t)m * H_ + col];
            h_out[(size_t)m * H_ + col] = (1.0f - z) * n + z * hp;
        }
    }
}

// -------- output projection: logits = h @ W_out^T + b_out --------
// 8 waves/block; h staged to LDS once per block (async-to-LDS), each wave
// computes TWO adjacent 16-col tiles of V (4 WMMAs / 2 vmem / 2 ds per K-step).
__global__ __launch_bounds__(256) void logits_kernel(
    const float* __restrict__ h,      // (B,H)
    const float* __restrict__ W_out,  // (V,H)
    const float* __restrict__ b_out,  // (V)
    float*       __restrict__ dec_out,// (B,L,V)
    int t)
{
    __shared__ float sh[B_ * H_];     // 128 KB staged h
    const int tid = threadIdx.x;

    {   // ---- async stage h (B x H) into LDS ----
        const unsigned lbase = (unsigned)(size_t)(void*)sh;
#pragma unroll 1
        for (int i = 0; i < (B_ * H_) / (256 * 4); ++i) {   // 32 iters
            const int f = (i * 256 + tid) * 4;
            async_copy_b128(lbase + (unsigned)f * 4u, (const void*)(h + f));
        }
        wait_async_zero();
        __syncthreads();
    }

    const int wave = tid >> 5;
    const int lane = tid & 31;
    const int half = lane >> 4;
    const int mlo  = lane & 15;
    const int koff = half * 2;
    const int n0   = (blockIdx.x * 8 + wave) * 32;  // two adjacent 16-col tiles

    const float* sh0 = sh + (size_t)mlo * H_ + koff;
    const float* sh1 = sh + (size_t)(16 + mlo) * H_ + koff;
    const float* pw0 = W_out + (size_t)(n0 + mlo) * H_ + koff;
    const float* pw1 = W_out + (size_t)(n0 + 16 + mlo) * H_ + koff;

    v8f acc[2][2] = {};   // [n-tile][m-tile]
#pragma unroll 4
    for (int k = 0; k < H_; k += 4) {
        v2f a0 = *(const v2f*)(sh0 + k);    // ds_load_b64
        v2f a1 = *(const v2f*)(sh1 + k);
        v2f b0 = *(const v2f*)(pw0 + k);    // global_load_b64 (L2 stream)
        v2f b1 = *(const v2f*)(pw1 + k);
        acc[0][0] = wmma_f32_k4(a0, b0, acc[0][0]);
        acc[0][1] = wmma_f32_k4(a1, b0, acc[0][1]);
        acc[1][0] = wmma_f32_k4(a0, b1, acc[1][0]);
        acc[1][1] = wmma_f32_k4(a1, b1, acc[1][1]);
    }

#pragma unroll
    for (int nt = 0; nt < 2; ++nt) {
        const int col = n0 + nt * 16 + mlo;
        const float bias = b_out[col];
#pragma unroll
        for (int tm = 0; tm < 2; ++tm) {
#pragma unroll
            for (int i = 0; i < 8; ++i) {
                const int m = tm * 16 + half * 8 + i;
                dec_out[((size_t)m * L_ + t) * (size_t)V_ + col] = acc[nt][tm][i] + bias;
            }
        }
    }
}

// -------- per-row softmax-max / argmax / score; emit next token --------
__global__ __launch_bounds__(256) void argmax_softmax_kernel(
    const float* __restrict__ dec_out,   // (B,L,V)
    float* __restrict__ out_tokens,      // (L,B) flat
    float* __restrict__ out_scores,      // (L,B) flat
    int*   __restrict__ tok_next,        // (B)
    int t)
{
    __shared__ float smax[256];
    __shared__ int   sidx[256];
    __shared__ float ssum[256];
    const int b   = blockIdx.x;
    const int tid = threadIdx.x;
    const float* row = dec_out + ((size_t)b * L_ + t) * (size_t)V_;

    float lmax = -INFINITY;
    int   lidx = 0;
    for (int v = tid; v < V_; v += 256) {
        float x = row[v];
        if (x > lmax) { lmax = x; lidx = v; }
    }
    smax[tid] = lmax; sidx[tid] = lidx;
    __syncthreads();
    for (int s = 128; s > 0; s >>= 1) {
        if (tid < s) {
            float o = smax[tid + s]; int oi = sidx[tid + s];
            if (o > smax[tid] || (o == smax[tid] && oi < sidx[tid])) {
                smax[tid] = o; sidx[tid] = oi;
            }
        }
        __syncthreads();
    }
    const float gmax = smax[0];

    float lsum = 0.0f;
    for (int v = tid; v < V_; v += 256) lsum += __expf(row[v] - gmax);
    ssum[tid] = lsum;
    __syncthreads();
    for (int s = 128; s > 0; s >>= 1) {
        if (tid < s) ssum[tid] += ssum[tid + s];
        __syncthreads();
    }

    if (tid == 0) {
        const int idx = sidx[0];
        out_tokens[(size_t)t * B_ + b] = (float)idx;      // greedy token
        out_scores[(size_t)t * B_ + b] = 1.0f / ssum[0];  // exp(max-max)/sumexp
        tok_next[b] = idx;
    }
}

extern "C" void kernel_launch(void* const* d_in, const int* in_sizes, int n_in,
                              void* d_out, int out_size, void* d_ws, size_t ws_size,
                              hipStream_t stream) {
    const int*   dec_input = (const int*)d_in[0];
    const float* embed     = (const float*)d_in[1];
    const float* W_ih      = (const float*)d_in[2];
    const float* W_hh      = (const float*)d_in[3];
    const float* b_ih      = (const float*)d_in[4];
    const float* b_hh      = (const float*)d_in[5];
    const float* W_out     = (const float*)d_in[6];
    const float* b_out     = (const float*)d_in[7];
    (void)in_sizes; (void)n_in; (void)out_size; (void)ws_size;

    float* out        = (float*)d_out;
    float* out_tokens = out;                         // (L*B)
    float* out_scores = out + (size_t)L_ * B_;       // (L*B)
    float* dec_out    = out + (size_t)2 * L_ * B_;   // (B,L,V)

    float* h_buf0  = (float*)d_ws;                   // (B,H)
    float* h_buf1  = h_buf0 + (size_t)B_ * H_;       // (B,H)
    int*   tok_buf0 = (int*)(h_buf1 + (size_t)B_ * H_);
    int*   tok_buf1 = tok_buf0 + B_;

    init_kernel<<<(B_ * H_ + 255) / 256, 256, 0, stream>>>(dec_input, h_buf0, tok_buf0);

    for (int t = 0; t < L_; ++t) {
        float* h_in  = (t & 1) ? h_buf1 : h_buf0;
        float* h_out = (t & 1) ? h_buf0 : h_buf1;
        int* tin   = (t & 1) ? tok_buf1 : tok_buf0;
        int* tnext = (t & 1) ? tok_buf0 : tok_buf1;

        gru_step_kernel<<<H_ / 64, 128, 0, stream>>>(
            tin, embed, W_ih, W_hh, b_ih, b_hh, h_in, h_out);
        logits_kernel<<<V_ / 256, 256, 0, stream>>>(
            h_out, W_out, b_out, dec_out, t);
        argmax_softmax_kernel<<<B_, 256, 0, stream>>>(
            dec_out, out_tokens, out_scores, tnext, t);
    }
}